// Router_46377056862301
// MI455X (gfx1250) — compile-verified
//
#include <hip/hip_runtime.h>
#include <hip/hip_bf16.h>

typedef __attribute__((ext_vector_type(2))) float v2f;
typedef __attribute__((ext_vector_type(4))) float v4f;
typedef __attribute__((ext_vector_type(8))) float v8f;

#define NTOK   4096
#define DDIM   1024
#define NEXP   8
#define CAP    1280
#define CB_ELEMS   (NTOK * NEXP * CAP)          /* 41,943,040 */
#define OUT_CB_OFF 8
#define OUT_MASK_OFF (8 + CB_ELEMS)
#define OUT_TOTAL  (8 + 2 * CB_ELEMS)           /* 83,886,088 floats */

/* ---------------- Phase 1: streaming zero-fill of the 335.5 MB output ----- */
__global__ void __launch_bounds__(256) fill_zero_kernel(float* __restrict__ out, long n4) {
    long i      = (long)blockIdx.x * blockDim.x + threadIdx.x;
    long stride = (long)gridDim.x * blockDim.x;
    v4f z = {0.f, 0.f, 0.f, 0.f};
    v4f* o4 = (v4f*)out;
    for (; i < n4; i += stride) {
        __builtin_nontemporal_store(z, o4 + i);
    }
}

/* ---------------- Phase 2: fp32 router GEMM via V_WMMA_F32_16X16X4_F32 ---- */
/* block = 256 threads = 8 waves; each wave computes a 16-token x 16-col tile
   (cols 8..15 are zero padding). grid = 4096/(8*16) = 32 blocks.            */
__global__ void __launch_bounds__(256) router_gemm_kernel(const float* __restrict__ x,
                                                          const float* __restrict__ wg,
                                                          float* __restrict__ logits) {
    /* B staged as [kpair][n] float2: wlds[kp*16+n] = {wg[n][2kp], wg[n][2kp+1]},
       zero for n >= 8.  512*16 float2 = 64 KB.                               */
    __shared__ v2f wlds[512 * 16];
    int tid = threadIdx.x;
    for (int i = tid; i < 512 * 16; i += 256) {
        int n  = i & 15;
        int kp = i >> 4;
        v2f v = {0.f, 0.f};
        if (n < NEXP) {
            v.x = wg[n * DDIM + 2 * kp];
            v.y = wg[n * DDIM + 2 * kp + 1];
        }
        wlds[i] = v;
    }
    __syncthreads();

    int wave = tid >> 5;
    int lane = tid & 31;
    int m    = lane & 15;   /* A: row M   / B: col N / D: col N */
    int kh   = lane >> 4;   /* K-half for A and B, M-half for D */
    int tb   = (blockIdx.x * 8 + wave) * 16;

    const float* xrow = x + (long)(tb + m) * DDIM + 2 * kh;

    v8f acc = {0.f, 0.f, 0.f, 0.f, 0.f, 0.f, 0.f, 0.f};
    #pragma unroll 4
    for (int kb = 0; kb < DDIM; kb += 4) {
        v2f a = *(const v2f*)(xrow + kb);
        v2f b = wlds[((kb >> 1) + kh) * 16 + m];
        acc = __builtin_amdgcn_wmma_f32_16x16x4_f32(
            /*neg_a=*/false, a, /*neg_b=*/false, b,
            /*c_mod=*/(short)0, acc, /*reuse_a=*/false, /*reuse_b=*/false);
    }

    /* D layout: VGPR r, lane L -> D[r + 8*(L>>4)][L&15] */
    if (m < NEXP) {
        #pragma unroll
        for (int r = 0; r < 8; ++r) {
            logits[(long)(tb + r + 8 * kh) * NEXP + m] = acc[r];
        }
    }
}

/* ---------------- Phase 3: per-token top-2 + 2-way softmax ---------------- */
__global__ void __launch_bounds__(256) top2_kernel(const float* __restrict__ logits,
                                                   int* __restrict__ e0, int* __restrict__ e1,
                                                   float* __restrict__ p0, float* __restrict__ p1) {
    int t = blockIdx.x * blockDim.x + threadIdx.x;
    if (t >= NTOK) return;
    const float* l = logits + (long)t * NEXP;
    float v1 = -INFINITY, v2 = -INFINITY;
    int   i1 = 0, i2 = 0;
    #pragma unroll
    for (int e = 0; e < NEXP; ++e) {
        float v = l[e];
        if (v > v1)      { v2 = v1; i2 = i1; v1 = v; i1 = e; }
        else if (v > v2) { v2 = v;  i2 = e; }
    }
    /* softmax over {v1, v2} shifted by max (v1): */
    float d = expf(v2 - v1);
    float s = 1.f / (1.f + d);
    e0[t] = i1; e1[t] = i2;
    p0[t] = s;  p1[t] = d * s;
}

/* ---------------- Phase 4: k-major rank scan + capacity drop + scatter ---- */
/* Single workgroup, 1024 threads = 32 waves.  Processes the 8192-assignment
   sequence (k*4096 + t) in 8 chunks of 1024, exactly matching the reference's
   cumsum order.  Wave32 ballot -> in-wave prefix; per-wave histograms in LDS;
   8 threads do the cross-wave exclusive scan with a running per-expert base. */
__global__ void __launch_bounds__(1024) scan_scatter_kernel(const int* __restrict__ e0,
                                                            const int* __restrict__ e1,
                                                            const float* __restrict__ p0,
                                                            const float* __restrict__ p1,
                                                            float* __restrict__ out) {
    __shared__ int hist[32][NEXP];
    __shared__ int base[NEXP];
    int tid  = threadIdx.x;
    int wid  = tid >> 5;
    int lane = tid & 31;

    if (tid < NEXP) base[tid] = 0;
    __syncthreads();

    for (int chunk = 0; chunk < 2 * NTOK; chunk += 1024) {
        int a = chunk + tid;
        int k = a >> 12;            /* 0: top-1 pass, 1: top-2 pass */
        int t = a & (NTOK - 1);
        int e   = k ? e1[t] : e0[t];
        float p = k ? p1[t] : p0[t];

        unsigned b[NEXP];
        #pragma unroll
        for (int E = 0; E < NEXP; ++E)
            b[E] = __builtin_amdgcn_ballot_w32(e == E);

        unsigned lt = (1u << lane) - 1u;
        int myprefix = __popc(b[e] & lt);

        if (lane == 0) {
            #pragma unroll
            for (int E = 0; E < NEXP; ++E) hist[wid][E] = __popc(b[E]);
        }
        __syncthreads();

        if (tid < NEXP) {
            int run = base[tid];
            for (int w = 0; w < 32; ++w) {
                int c = hist[w][tid];
                hist[w][tid] = run;
                run += c;
            }
            base[tid] = run;
        }
        __syncthreads();

        int rank = hist[wid][e] + myprefix;
        if (rank < CAP) {
            long idx = (long)t * (NEXP * CAP) + (long)e * CAP + rank;
            out[OUT_CB_OFF + idx]   = p;
            out[OUT_MASK_OFF + idx] = (p != 0.f) ? 1.f : 0.f;
        }
        __syncthreads();   /* hist reused next chunk */
    }

    if (tid < NEXP) {
        int tot = base[tid];
        out[tid] = (float)(tot < CAP ? tot : CAP);
    }
}

extern "C" void kernel_launch(void* const* d_in, const int* in_sizes, int n_in,
                              void* d_out, int out_size, void* d_ws, size_t ws_size,
                              hipStream_t stream) {
    const float* x  = (const float*)d_in[0];   /* [2,2048,1024] fp32 */
    const float* wg = (const float*)d_in[1];   /* [8,1024] fp32      */
    float* out = (float*)d_out;
    float* ws  = (float*)d_ws;

    float* logits = ws;                            /* 4096*8 floats  */
    int*   e0     = (int*)(ws + NTOK * NEXP);      /* 4096 ints      */
    int*   e1     = e0 + NTOK;
    float* p0     = (float*)(e1 + NTOK);
    float* p1     = p0 + NTOK;

    long n4 = (long)OUT_TOTAL / 4;                 /* 20,971,522 float4 */
    hipLaunchKernelGGL(fill_zero_kernel,   dim3(4096), dim3(256), 0, stream, out, n4);
    hipLaunchKernelGGL(router_gemm_kernel, dim3(32),   dim3(256), 0, stream, x, wg, logits);
    hipLaunchKernelGGL(top2_kernel,        dim3(16),   dim3(256), 0, stream, logits, e0, e1, p0, p1);
    hipLaunchKernelGGL(scan_scatter_kernel, dim3(1),   dim3(1024), 0, stream, e0, e1, p0, p1, out);
}